// Geodesic_GNN_55044300866303
// MI455X (gfx1250) — compile-verified
//
#include <hip/hip_runtime.h>
#include <math.h>

// ---------------------------------------------------------------------------
// Geodesic GNN layer for MI455X (gfx1250, wave32, WMMA f32 16x16x4)
// All GEMM-like work (edge pipeline AND node GVLinears) runs on
// V_WMMA_F32_16X16X4_F32 with 16-item batches staged in LDS.
// LDS row stride padded to 18 floats to avoid bank conflicts between the
// two half-waves of WMMA D-stores (8*18 = 16 mod 64 banks).
// All helpers force-inlined so infer-address-spaces keeps LDS accesses as
// ds_* (no generic/flat pointers) and weight loads as global_load_*.
// ---------------------------------------------------------------------------

typedef float v2f __attribute__((ext_vector_type(2)));
typedef float v8f __attribute__((ext_vector_type(8)));

#define LROW 18
__device__ __forceinline__ int lidx(int r, int c) { return r * LROW + c; }

__device__ __forceinline__ float sigm(float x) { return 1.0f / (1.0f + __expf(-x)); }

// Native CDNA5 non-returning float atomic (STOREcnt-tracked; s_endpgm waits idle)
__device__ __forceinline__ void atomic_add_f32(float* p, float v)
{
    asm volatile("global_atomic_add_f32 %0, %1, off" : : "v"(p), "v"(v) : "memory");
}

// ---------------------------------------------------------------------------
// Wave-level GEMM:  OUT[OD x 16] = W[OD x ID] @ X[ID x 16]  (+ bias[OD])
//   W    : global, row-major (uniform across waves -> L2 resident)
//   X    : LDS, padded row-major  X[k*LROW + item]  (== WMMA B layout)
//   OUT  : LDS, padded row-major  OUT[o*LROW + item]
// OD in {16,32,64}, ID multiple of 4. Full wave required (EXEC all ones).
// ---------------------------------------------------------------------------
__device__ __forceinline__ void wave_gemm(const float* __restrict__ W, int ID,
                                          const float* X, float* OUT, int OD,
                                          const float* __restrict__ bias)
{
    const int lane = threadIdx.x & 31;
    const int m    = lane & 15;       // row (A) / col (B,D)
    const int hi   = lane >> 4;       // half-wave select
    const int kb   = hi << 1;         // K sub-offset: lanes 16-31 hold K=2,3

    for (int t = 0; t < OD; t += 16) {
        v8f acc = {0.f, 0.f, 0.f, 0.f, 0.f, 0.f, 0.f, 0.f};
        const float* wr = W + (size_t)(t + m) * ID + kb;
        for (int k0 = 0; k0 < ID; k0 += 4) {
            v2f a, b;
            a.x = wr[k0];
            a.y = wr[k0 + 1];
            b.x = X[(k0 + kb) * LROW + m];
            b.y = X[(k0 + kb + 1) * LROW + m];
            acc = __builtin_amdgcn_wmma_f32_16x16x4_f32(false, a, false, b,
                                                        (short)0, acc, false, false);
        }
#pragma unroll
        for (int j = 0; j < 8; ++j) {
            int r = t + j + (hi << 3);
            float v = acc[j];
            if (bias) v += bias[r];
            OUT[r * LROW + m] = v;
        }
    }
}

struct GvW { const float *Wv1, *Wv2, *Ws, *Wg, *bg; };

// ---------------------------------------------------------------------------
// Wave GVLinear for the (64 sca, 16 vec) -> (64 sca, 16 vec) shape, batched
// over 16 items (nodes). X1 rows 16..79 must hold input sca; VD* hold input
// vec per spatial dim. Produces OS (out_sca 64x16) and gated OV* (16x16 x3).
// Destroys X1 rows 0..15, VI*, GL.  FORCE-INLINED: keeps LDS in addrspace(3).
// ---------------------------------------------------------------------------
__device__ __forceinline__ void gv_wave(GvW p, float* X1,
                        const float* VD0, const float* VD1, const float* VD2,
                        float* VI0, float* VI1, float* VI2,
                        float* OS, float* GL, float* OV0, float* OV1, float* OV2)
{
    const int lane = threadIdx.x & 31;
    wave_gemm(p.Wv1, 16, VD0, VI0, 16, nullptr);
    wave_gemm(p.Wv1, 16, VD1, VI1, 16, nullptr);
    wave_gemm(p.Wv1, 16, VD2, VI2, 16, nullptr);
    __syncthreads();
    for (int i = lane; i < 16 * 16; i += 32) {          // v_norm -> X1 rows 0..15
        int r = i >> 4, c = i & 15;
        float a = VI0[lidx(r, c)], b = VI1[lidx(r, c)], d = VI2[lidx(r, c)];
        X1[lidx(r, c)] = sqrtf(a * a + b * b + d * d);
    }
    __syncthreads();
    wave_gemm(p.Ws, 80, X1, OS, 64, nullptr);           // out_sca
    __syncthreads();
    wave_gemm(p.Wg, 64, OS, GL, 16, p.bg);              // gate pre-act
    __syncthreads();
    for (int i = lane; i < 16 * 16; i += 32) {
        int r = i >> 4, c = i & 15;
        GL[lidx(r, c)] = sigm(GL[lidx(r, c)]);
    }
    __syncthreads();
    wave_gemm(p.Wv2, 16, VI0, OV0, 16, nullptr);
    wave_gemm(p.Wv2, 16, VI1, OV1, 16, nullptr);
    wave_gemm(p.Wv2, 16, VI2, OV2, 16, nullptr);
    __syncthreads();
    for (int i = lane; i < 16 * 16; i += 32) {          // gate the vectors
        int r = i >> 4, c = i & 15;
        float g = GL[lidx(r, c)];
        OV0[lidx(r, c)] *= g; OV1[lidx(r, c)] *= g; OV2[lidx(r, c)] *= g;
    }
    __syncthreads();
}

// ---------------------------------------------------------------------------
// Kernel 0: fold the rank-1 edge_vec through edge_mapper's Wv1/Wv2.
// ---------------------------------------------------------------------------
__global__ void prep_kernel(const float* __restrict__ emWv1,
                            const float* __restrict__ emWv2,
                            const float* __restrict__ wexp,
                            float* __restrict__ w1e, float* __restrict__ w2e)
{
    if (threadIdx.x == 0 && blockIdx.x == 0) {
        float t1[16];
        for (int h = 0; h < 16; ++h) {
            float a = 0.f;
            for (int v = 0; v < 16; ++v) a += emWv1[h * 16 + v] * wexp[v];
            t1[h] = a; w1e[h] = a;
        }
        for (int o = 0; o < 16; ++o) {
            float a = 0.f;
            for (int h = 0; h < 16; ++h) a += emWv2[o * 16 + h] * t1[h];
            w2e[o] = a;
        }
    }
}

// ---------------------------------------------------------------------------
// Kernel 1 (16 nodes per wave, WMMA): resi + node_mapper GVLinears, then fold
// node_mapper outputs through nss/nsv/nvv; zero-init aggregation buffers.
// ---------------------------------------------------------------------------
struct NodeArgs {
    const float *node_sca, *node_vec;
    GvW nm, resi;
    const float *nssW, *nssB, *nsvW, *nsvB, *nvvW;
    float *nss, *nsv, *nvv, *rsca, *rvec, *asca, *avec;
    int N;
};

__global__ __launch_bounds__(32) void node_pre_kernel(NodeArgs A)
{
    __shared__ float X1[80 * LROW];
    __shared__ float VD[3][16 * LROW];
    __shared__ float VI[3][16 * LROW];
    __shared__ float OS[64 * LROW];
    __shared__ float GL[16 * LROW];
    __shared__ float OV[3][16 * LROW];
    __shared__ float NS[32 * LROW];

    const int lane = threadIdx.x;
    const int n0   = blockIdx.x * 16;

    // ---- stage inputs (clamped gather) ------------------------------------
    for (int i = lane; i < 64 * 16; i += 32) {
        int r = i >> 4, c = i & 15;
        int n = min(n0 + c, A.N - 1);
        X1[lidx(16 + r, c)] = A.node_sca[(size_t)n * 64 + r];
    }
    for (int d = 0; d < 3; ++d)
        for (int i = lane; i < 16 * 16; i += 32) {
            int r = i >> 4, c = i & 15;
            int n = min(n0 + c, A.N - 1);
            VD[d][lidx(r, c)] = A.node_vec[(size_t)n * 48 + r * 3 + d];
        }
    __syncthreads();

    // ---- residual branch ---------------------------------------------------
    gv_wave(A.resi, X1, VD[0], VD[1], VD[2], VI[0], VI[1], VI[2],
            OS, GL, OV[0], OV[1], OV[2]);
    for (int i = lane; i < 64 * 16; i += 32) {
        int r = i >> 4, c = i & 15; int n = n0 + c;
        if (n < A.N) A.rsca[(size_t)n * 64 + r] = OS[lidx(r, c)];
    }
    for (int d = 0; d < 3; ++d)
        for (int i = lane; i < 16 * 16; i += 32) {
            int r = i >> 4, c = i & 15; int n = n0 + c;
            if (n < A.N) A.rvec[(size_t)n * 48 + r * 3 + d] = OV[d][lidx(r, c)];
        }
    __syncthreads();

    // ---- node_mapper branch (X1 rows 16..79 still hold sca) ----------------
    gv_wave(A.nm, X1, VD[0], VD[1], VD[2], VI[0], VI[1], VI[2],
            OS, GL, OV[0], OV[1], OV[2]);

    // nss / nsv from n_sca (OS)
    wave_gemm(A.nssW, 64, OS, NS, 32, A.nssB);
    __syncthreads();
    for (int i = lane; i < 32 * 16; i += 32) {
        int r = i >> 4, c = i & 15; int n = n0 + c;
        if (n < A.N) A.nss[(size_t)n * 32 + r] = NS[lidx(r, c)];
    }
    __syncthreads();
    wave_gemm(A.nsvW, 64, OS, NS, 32, A.nsvB);
    __syncthreads();
    for (int i = lane; i < 32 * 16; i += 32) {
        int r = i >> 4, c = i & 15; int n = n0 + c;
        if (n < A.N) A.nsv[(size_t)n * 32 + r] = NS[lidx(r, c)];
    }
    __syncthreads();
    // nvv from n_vec (gated OV), per spatial dim
    for (int d = 0; d < 3; ++d) {
        wave_gemm(A.nvvW, 16, OV[d], NS, 32, nullptr);
        __syncthreads();
        for (int i = lane; i < 32 * 16; i += 32) {
            int r = i >> 4, c = i & 15; int n = n0 + c;
            if (n < A.N) A.nvv[(size_t)n * 96 + r * 3 + d] = NS[lidx(r, c)];
        }
        __syncthreads();
    }
    // zero aggregation buffers
    for (int i = lane; i < 64 * 16; i += 32) {
        int r = i >> 4, c = i & 15; int n = n0 + c;
        if (n < A.N) A.asca[(size_t)n * 64 + r] = 0.f;
    }
    for (int i = lane; i < 48 * 16; i += 32) {
        int r = i >> 4, c = i & 15; int n = n0 + c;
        if (n < A.N) A.avec[(size_t)n * 48 + r] = 0.f;
    }
}

// ---------------------------------------------------------------------------
// Kernel 2 (16 edges per wave, WMMA): full edge pipeline + scatter-add.
// ---------------------------------------------------------------------------
struct EdgeArgs {
    const float *pos, *efeat, *gds;
    const int   *eidx;
    const float *emWs, *emWg, *embg;
    const float *essW, *essB, *esvW, *esvB, *evvW;
    const float *moWv1, *moWv2, *moWs, *moWg, *mobg;
    const float *w1e, *w2e;
    const float *nss, *nsv, *nvv;
    float *asca, *avec;
    int E;
};

__global__ __launch_bounds__(32) void edge_kernel(EdgeArgs A)
{
    __shared__ float sX1[32 * LROW];     // [vnorm(16) ; edge_sca(16)]
    __shared__ float sESCA[64 * LROW];   // edge_mapper out_sca
    __shared__ float sGL[16 * LROW];     // gate pre-activations (reused)
    __shared__ float sG2[16 * LROW];     // sigmoid(gate)*w2e
    __shared__ float sESS[32 * LROW];
    __shared__ float sESV[32 * LROW];
    __shared__ float sEG[32 * LROW];     // evv_W @ G2
    __shared__ float sNSV[32 * LROW];    // gathered nsv
    __shared__ float sMV[32 * LROW];     // msg_vec one dim
    __shared__ float sVI[3][32 * LROW];  // msg_out v_inter, 3 dims
    __shared__ float sX2[64 * LROW];     // [vnorm(32) ; msg_sca(32)]
    __shared__ float sOS[64 * LROW];     // msg_out out_sca
    __shared__ float sOV[16 * LROW];     // msg_out out_vec one dim
    __shared__ int   sRow[16];
    __shared__ float sCoeff[16];
    __shared__ float sEv[3][16];

    const int lane = threadIdx.x;
    const int e0   = blockIdx.x * 16;

    // ---- stage 0: per-edge scalars (one edge per lane, lanes 0..15) -------
    if (lane < 16) {
        int e   = min(e0 + lane, A.E - 1);
        int row = A.eidx[e];
        int col = A.eidx[A.E + e];
        sRow[lane] = row;
        float dx = A.pos[(size_t)row * 3 + 0] - A.pos[(size_t)col * 3 + 0];
        float dy = A.pos[(size_t)row * 3 + 1] - A.pos[(size_t)col * 3 + 1];
        float dz = A.pos[(size_t)row * 3 + 2] - A.pos[(size_t)col * 3 + 2];
        float nrm = sqrtf(dx * dx + dy * dy + dz * dz);
        float inv = 1.0f / (nrm + 1e-7f);
        sEv[0][lane] = dx * inv; sEv[1][lane] = dy * inv; sEv[2][lane] = dz * inv;
        float evlen = nrm * inv;
        float dist  = A.gds[e];
        float co = 0.5f * (__cosf(dist * (3.14159265358979323846f / 10.0f)) + 1.0f);
        co = (dist <= 10.0f && dist >= 0.0f) ? co : 0.0f;
        sCoeff[lane] = co;
#pragma unroll
        for (int h = 0; h < 16; ++h)
            sX1[lidx(h, lane)] = fabsf(A.w1e[h]) * evlen;
        const float step = 10.0f / 13.0f;               // linspace(0,10,14)
        const float gco  = -0.5f / (step * step);
#pragma unroll
        for (int k = 0; k < 14; ++k) {
            float t = dist - (float)k * step;
            sX1[lidx(16 + k, lane)] = __expf(gco * t * t);
        }
        sX1[lidx(16 + 14, lane)] = A.efeat[(size_t)e * 2 + 0];
        sX1[lidx(16 + 15, lane)] = A.efeat[(size_t)e * 2 + 1];
    }
    __syncthreads();

    // ---- edge_mapper scalar path + mixing matmuls (WMMA) ------------------
    wave_gemm(A.emWs, 32, sX1, sESCA, 64, nullptr);
    __syncthreads();
    wave_gemm(A.emWg, 64, sESCA, sGL, 16, A.embg);
    wave_gemm(A.essW, 64, sESCA, sESS, 32, A.essB);
    wave_gemm(A.esvW, 64, sESCA, sESV, 32, A.esvB);
    __syncthreads();

    for (int i = lane; i < 16 * 16; i += 32) {
        int o = i >> 4, j = i & 15;
        sG2[lidx(o, j)] = sigm(sGL[lidx(o, j)]) * A.w2e[o];
    }
    __syncthreads();
    wave_gemm(A.evvW, 16, sG2, sEG, 32, nullptr);
    __syncthreads();

    // ---- gather per-node precomputes, build msg_sca -----------------------
    for (int i = lane; i < 32 * 16; i += 32) {
        int h = i >> 4, j = i & 15;
        int r = sRow[j];
        sX2[lidx(32 + h, j)] = A.nss[(size_t)r * 32 + h] * sESS[lidx(h, j)] * sCoeff[j];
        sNSV[lidx(h, j)] = A.nsv[(size_t)r * 32 + h];
    }
    __syncthreads();

    // ---- msg_vec per spatial dim + msg_out Wv1 ----------------------------
    for (int d = 0; d < 3; ++d) {
        for (int i = lane; i < 32 * 16; i += 32) {
            int h = i >> 4, j = i & 15;
            float nvv_v = A.nvv[(size_t)sRow[j] * 96 + h * 3 + d];
            float v1 = nvv_v * sESV[lidx(h, j)];
            float v2 = sNSV[lidx(h, j)] * sEG[lidx(h, j)] * sEv[d][j];
            sMV[lidx(h, j)] = (v1 + v2) * sCoeff[j];
        }
        __syncthreads();
        wave_gemm(A.moWv1, 32, sMV, &sVI[d][0], 32, nullptr);
        __syncthreads();
    }

    // ---- msg_out: v_norm, Ws, Wg ------------------------------------------
    for (int i = lane; i < 32 * 16; i += 32) {
        int h = i >> 4, j = i & 15;
        float a = sVI[0][lidx(h, j)], b = sVI[1][lidx(h, j)], c = sVI[2][lidx(h, j)];
        sX2[lidx(h, j)] = sqrtf(a * a + b * b + c * c);
    }
    __syncthreads();
    wave_gemm(A.moWs, 64, sX2, sOS, 64, nullptr);
    __syncthreads();
    wave_gemm(A.moWg, 64, sOS, sGL, 16, A.mobg);
    __syncthreads();
    for (int i = lane; i < 16 * 16; i += 32) {
        int o = i >> 4, j = i & 15;
        sGL[lidx(o, j)] = sigm(sGL[lidx(o, j)]);
    }
    __syncthreads();

    // ---- scatter-add aggregation (native f32 atomics into L2) -------------
    for (int i = lane; i < 64 * 16; i += 32) {
        int o = i >> 4, j = i & 15;
        if (e0 + j < A.E)
            atomic_add_f32(&A.asca[(size_t)sRow[j] * 64 + o], sOS[lidx(o, j)]);
    }
    for (int d = 0; d < 3; ++d) {
        wave_gemm(A.moWv2, 32, &sVI[d][0], sOV, 16, nullptr);
        __syncthreads();
        for (int i = lane; i < 16 * 16; i += 32) {
            int o = i >> 4, j = i & 15;
            if (e0 + j < A.E)
                atomic_add_f32(&A.avec[(size_t)sRow[j] * 48 + o * 3 + d],
                               sGL[lidx(o, j)] * sOV[lidx(o, j)]);
        }
        __syncthreads();
    }
}

// ---------------------------------------------------------------------------
// Kernel 3 (16 nodes per wave, WMMA): out = aggr_gv(r + aggr)
// ---------------------------------------------------------------------------
struct PostArgs {
    GvW ag;
    const float *rsca, *rvec, *asca, *avec;
    float *out;
    int N;
};

__global__ __launch_bounds__(32) void node_post_kernel(PostArgs A)
{
    __shared__ float X1[80 * LROW];
    __shared__ float VD[3][16 * LROW];
    __shared__ float VI[3][16 * LROW];
    __shared__ float OS[64 * LROW];
    __shared__ float GL[16 * LROW];
    __shared__ float OV[3][16 * LROW];

    const int lane = threadIdx.x;
    const int n0   = blockIdx.x * 16;

    for (int i = lane; i < 64 * 16; i += 32) {
        int r = i >> 4, c = i & 15;
        int n = min(n0 + c, A.N - 1);
        X1[lidx(16 + r, c)] = A.rsca[(size_t)n * 64 + r] + A.asca[(size_t)n * 64 + r];
    }
    for (int d = 0; d < 3; ++d)
        for (int i = lane; i < 16 * 16; i += 32) {
            int r = i >> 4, c = i & 15;
            int n = min(n0 + c, A.N - 1);
            VD[d][lidx(r, c)] = A.rvec[(size_t)n * 48 + r * 3 + d]
                              + A.avec[(size_t)n * 48 + r * 3 + d];
        }
    __syncthreads();

    gv_wave(A.ag, X1, VD[0], VD[1], VD[2], VI[0], VI[1], VI[2],
            OS, GL, OV[0], OV[1], OV[2]);

    for (int i = lane; i < 64 * 16; i += 32) {
        int r = i >> 4, c = i & 15; int n = n0 + c;
        if (n < A.N) A.out[(size_t)n * 64 + r] = OS[lidx(r, c)];
    }
    float* ovp = A.out + (size_t)A.N * 64;
    for (int d = 0; d < 3; ++d)
        for (int i = lane; i < 16 * 16; i += 32) {
            int r = i >> 4, c = i & 15; int n = n0 + c;
            if (n < A.N) ovp[(size_t)n * 48 + r * 3 + d] = OV[d][lidx(r, c)];
        }
}

// ---------------------------------------------------------------------------
extern "C" void kernel_launch(void* const* d_in, const int* in_sizes, int n_in,
                              void* d_out, int out_size, void* d_ws, size_t ws_size,
                              hipStream_t stream)
{
    const float* node_sca  = (const float*)d_in[0];
    const float* node_vec  = (const float*)d_in[1];
    const float* node_pos  = (const float*)d_in[2];
    const float* efeat     = (const float*)d_in[3];
    const int*   eidx      = (const int*)d_in[4];
    const float* gds       = (const float*)d_in[5];
    const float* Wexp      = (const float*)d_in[6];
    GvW nm   { (const float*)d_in[7],  (const float*)d_in[8],  (const float*)d_in[9],
               (const float*)d_in[10], (const float*)d_in[11] };
    const float* emWv1 = (const float*)d_in[12];
    const float* emWv2 = (const float*)d_in[13];
    const float* emWs  = (const float*)d_in[14];
    const float* emWg  = (const float*)d_in[15];
    const float* embg  = (const float*)d_in[16];
    const float* moWv1 = (const float*)d_in[17];
    const float* moWv2 = (const float*)d_in[18];
    const float* moWs  = (const float*)d_in[19];
    const float* moWg  = (const float*)d_in[20];
    const float* mobg  = (const float*)d_in[21];
    GvW resi { (const float*)d_in[22], (const float*)d_in[23], (const float*)d_in[24],
               (const float*)d_in[25], (const float*)d_in[26] };
    GvW ag   { (const float*)d_in[27], (const float*)d_in[28], (const float*)d_in[29],
               (const float*)d_in[30], (const float*)d_in[31] };
    const float* essW = (const float*)d_in[32];
    const float* essB = (const float*)d_in[33];
    const float* nssW = (const float*)d_in[34];
    const float* nssB = (const float*)d_in[35];
    const float* esvW = (const float*)d_in[36];
    const float* esvB = (const float*)d_in[37];
    const float* nsvW = (const float*)d_in[38];
    const float* nsvB = (const float*)d_in[39];
    const float* evvW = (const float*)d_in[40];
    const float* nvvW = (const float*)d_in[41];

    const int N = in_sizes[0] / 64;
    const int E = in_sizes[5];

    // workspace layout (floats)
    float* ws   = (float*)d_ws;
    float* w1e  = ws;
    float* w2e  = w1e + 16;
    float* nss  = w2e + 16;
    float* nsv  = nss + (size_t)N * 32;
    float* nvv  = nsv + (size_t)N * 32;
    float* rsca = nvv + (size_t)N * 96;
    float* rvec = rsca + (size_t)N * 64;
    float* asca = rvec + (size_t)N * 48;
    float* avec = asca + (size_t)N * 64;

    prep_kernel<<<1, 32, 0, stream>>>(emWv1, emWv2, Wexp, w1e, w2e);

    NodeArgs na;
    na.node_sca = node_sca; na.node_vec = node_vec;
    na.nm = nm; na.resi = resi;
    na.nssW = nssW; na.nssB = nssB; na.nsvW = nsvW; na.nsvB = nsvB; na.nvvW = nvvW;
    na.nss = nss; na.nsv = nsv; na.nvv = nvv;
    na.rsca = rsca; na.rvec = rvec; na.asca = asca; na.avec = avec;
    na.N = N;
    node_pre_kernel<<<(N + 15) / 16, 32, 0, stream>>>(na);

    EdgeArgs ea;
    ea.pos = node_pos; ea.efeat = efeat; ea.gds = gds; ea.eidx = eidx;
    ea.emWs = emWs; ea.emWg = emWg; ea.embg = embg;
    ea.essW = essW; ea.essB = essB; ea.esvW = esvW; ea.esvB = esvB; ea.evvW = evvW;
    ea.moWv1 = moWv1; ea.moWv2 = moWv2; ea.moWs = moWs; ea.moWg = moWg; ea.mobg = mobg;
    ea.w1e = w1e; ea.w2e = w2e;
    ea.nss = nss; ea.nsv = nsv; ea.nvv = nvv;
    ea.asca = asca; ea.avec = avec;
    ea.E = E;
    edge_kernel<<<(E + 15) / 16, 32, 0, stream>>>(ea);

    PostArgs pa;
    pa.ag = ag;
    pa.rsca = rsca; pa.rvec = rvec; pa.asca = asca; pa.avec = avec;
    pa.out = (float*)d_out; pa.N = N;
    node_post_kernel<<<(N + 15) / 16, 32, 0, stream>>>(pa);
}